// LightHeteroGNN_30030411334245
// MI455X (gfx1250) — compile-verified
//
#include <hip/hip_runtime.h>
#include <hip/hip_bf16.h>

typedef __attribute__((ext_vector_type(2))) float v2f;
typedef __attribute__((ext_vector_type(8))) float v8f;

#define H48 48
#define LN_EPS 1e-5f

#define N_ORDER  500000
#define N_DEVICE 100000
#define N_TYPE   200

__device__ __forceinline__ float elu_f(float x) {
    return x > 0.f ? x : (expf(x) - 1.f);
}

// ---------------- zero fill ----------------
__global__ void k_zero(float* __restrict__ p, long n) {
    long i = (long)blockIdx.x * blockDim.x + threadIdx.x;
    if (i < n) p[i] = 0.f;
}

// ---------------- encoder: out = elu(x @ W + b), W is [Fin,48] -------------
__global__ void k_enc(const float* __restrict__ x, const float* __restrict__ W,
                      const float* __restrict__ b, float* __restrict__ out,
                      int N, int Fin) {
    int i = blockIdx.x * blockDim.x + threadIdx.x;
    if (i >= N * H48) return;
    int n = i / H48, h = i - n * H48;
    float acc = b[h];
    for (int k = 0; k < Fin; ++k) acc += x[n * Fin + k] * W[k * H48 + h];
    out[i] = elu_f(acc);
}

// ---------------- per-destination edge counts ----------------
__global__ void k_count(const int* __restrict__ ei, int E, float* __restrict__ cnt) {
    int e = blockIdx.x * blockDim.x + threadIdx.x;
    if (e < E) atomicAdd(&cnt[ei[E + e]], 1.f);
}

__global__ void k_inv(float* __restrict__ c, int n) {
    int i = blockIdx.x * blockDim.x + threadIdx.x;
    if (i < n) c[i] = 1.f / fmaxf(c[i], 1.f);
}

// ---------------- weighted scatter: agg[dst] += feat[src] / cnt[dst] -------
__global__ void k_scatter(const float* __restrict__ feat, const int* __restrict__ ei,
                          const float* __restrict__ cinv, float* __restrict__ agg,
                          int E) {
    long i = (long)blockIdx.x * blockDim.x + threadIdx.x;
    if (i >= (long)E * H48) return;
    int e = (int)(i / H48);
    int h = (int)(i - (long)e * H48);
    int s = ei[e];
    int d = ei[E + e];
    atomicAdd(&agg[(long)d * H48 + h], feat[(long)s * H48 + h] * cinv[d]);
}

// ---------------- fused update MLP + LayerNorm via V_WMMA_F32_16X16X4_F32 --
// block = 96 threads = 3 waves; wave w handles rows [blk*48 + w*16, +16).
// A tile staged in LDS (b128 loads); W staged TRANSPOSED in LDS so both A and
// B fragments are contiguous, aligned ds_load_b64s straight into WMMA operand
// register pairs. Row stride K+4 is bank-conflict-free and 16B-aligned.
// The A tile region is reused as LayerNorm scratch afterwards.
template <int K>
__global__ void __launch_bounds__(96) k_update(
    const float* __restrict__ f0, const float* __restrict__ f1,
    const float* __restrict__ f2, const float* __restrict__ W,
    const float* __restrict__ b,  const float* __restrict__ g,
    const float* __restrict__ be, float* __restrict__ out, int N) {
    constexpr int STR = K + 4;              // LDS row stride (floats)
    __shared__ float sWT[48 * STR];         // W transposed: sWT[col][k]
    __shared__ float sA[3 * 16 * STR];      // per-wave A tile / LN scratch

    const int tid  = threadIdx.x;
    const int lane = tid & 31;
    const int wave = tid >> 5;
    const int half = lane >> 4;             // 0: lanes 0-15, 1: lanes 16-31
    const int l16  = lane & 15;

    // stage W transposed (shared by all 3 waves): coalesced global reads
    for (int i = tid; i < K * 48; i += 96) {
        const int k = i / 48, col = i - k * 48;
        sWT[col * STR + k] = W[i];
    }

    // stage this wave's 16-row concat A tile: [16, K] via float4
    const int mbase = blockIdx.x * 48 + wave * 16;
    float* myA = &sA[wave * 16 * STR];
    const float* fs[3] = { f0, f1, f2 };
    #pragma unroll
    for (int s = 0; s < K / 48; ++s) {
        const float* f = fs[s];
        #pragma unroll
        for (int it = 0; it < 6; ++it) {            // 16*12 quads / 32 lanes
            const int idx = it * 32 + lane;
            const int r = idx / 12, q = idx - r * 12;
            const long rg = (long)min(mbase + r, N - 1);
            const float4 v = *(const float4*)&f[rg * 48 + q * 4];
            *(float4*)&myA[r * STR + s * 48 + q * 4] = v;
        }
    }
    __syncthreads();

    // K-loop: A and B both contiguous b64 LDS loads, 3 N-tiles of 16
    const float* aRow = &myA[l16 * STR];
    const float* bR0  = &sWT[l16 * STR];          // cols  0..15
    const float* bR1  = &sWT[(l16 + 16) * STR];   // cols 16..31
    const float* bR2  = &sWT[(l16 + 32) * STR];   // cols 32..47
    v8f c0 = {}, c1 = {}, c2 = {};
    #pragma unroll
    for (int k = 0; k < K; k += 4) {
        const int ka = k + 2 * half;   // lanes 0-15: K=k,k+1; lanes 16-31: k+2,k+3
        v2f a, b0, b1, b2;
        a.x  = aRow[ka];  a.y  = aRow[ka + 1];
        b0.x = bR0[ka];   b0.y = bR0[ka + 1];
        b1.x = bR1[ka];   b1.y = bR1[ka + 1];
        b2.x = bR2[ka];   b2.y = bR2[ka + 1];
        c0 = __builtin_amdgcn_wmma_f32_16x16x4_f32(false, a, false, b0, (short)0, c0, false, false);
        c1 = __builtin_amdgcn_wmma_f32_16x16x4_f32(false, a, false, b1, (short)0, c1, false, false);
        c2 = __builtin_amdgcn_wmma_f32_16x16x4_f32(false, a, false, b2, (short)0, c2, false, false);
    }
    __syncthreads();

    // epilogue: bias + ELU into LDS (row m = r + 8*half, col = tile*16 + l16)
    #pragma unroll
    for (int r = 0; r < 8; ++r) {
        const int m = r + 8 * half;
        myA[m * STR + l16]      = elu_f(c0[r] + b[l16]);
        myA[m * STR + l16 + 16] = elu_f(c1[r] + b[l16 + 16]);
        myA[m * STR + l16 + 32] = elu_f(c2[r] + b[l16 + 32]);
    }
    __syncthreads();

    // LayerNorm: each lane owns row (lane&15); lane-half splits the 48 cols
    const int m = l16;
    float mu = 0.f;
    #pragma unroll
    for (int c = 0; c < 48; ++c) mu += myA[m * STR + c];
    mu *= (1.f / 48.f);
    float var = 0.f;
    #pragma unroll
    for (int c = 0; c < 48; ++c) {
        float d = myA[m * STR + c] - mu;
        var += d * d;
    }
    var *= (1.f / 48.f);
    const float rs = rsqrtf(var + LN_EPS);
    const int rg = mbase + m;
    if (rg < N) {
        const int cst = half * 24;
        for (int c = cst; c < cst + 24; ++c)
            out[(long)rg * 48 + c] = (myA[m * STR + c] - mu) * rs * g[c] + be[c];
    }
}

// ---------------------------------------------------------------------------
extern "C" void kernel_launch(void* const* d_in, const int* in_sizes, int n_in,
                              void* d_out, int out_size, void* d_ws, size_t ws_size,
                              hipStream_t stream) {
    const float* x_order  = (const float*)d_in[0];
    const float* x_device = (const float*)d_in[1];
    const float* x_type   = (const float*)d_in[2];
    const int* ei_do = (const int*)d_in[3];  const int E_do = in_sizes[3] / 2;
    const int* ei_to = (const int*)d_in[4];  const int E_to = in_sizes[4] / 2;
    const int* ei_od = (const int*)d_in[5];  const int E_od = in_sizes[5] / 2;
    const int* ei_dd = (const int*)d_in[6];  const int E_dd = in_sizes[6] / 2;
    const int* ei_td = (const int*)d_in[7];  const int E_td = in_sizes[7] / 2;
    const float* Wo  = (const float*)d_in[8];   const float* bo  = (const float*)d_in[9];
    const float* Wd  = (const float*)d_in[10];  const float* bd  = (const float*)d_in[11];
    const float* Wt  = (const float*)d_in[12];  const float* bt  = (const float*)d_in[13];
    const float* Wuo = (const float*)d_in[14];  const float* buo = (const float*)d_in[15];
    const float* Wud = (const float*)d_in[16];  const float* bud = (const float*)d_in[17];
    const float* g_o  = (const float*)d_in[18]; const float* be_o = (const float*)d_in[19];
    const float* g_d  = (const float*)d_in[20]; const float* be_d = (const float*)d_in[21];

    // workspace layout (floats)
    float* ws     = (float*)d_ws;
    float* ho     = ws;                         // 500000*48
    float* hd     = ho     + 24000000L;         // 100000*48
    float* ht     = hd     + 4800000L;          // 200*48
    float* aggO   = ht     + 9600L;             // mean(d->o) + mean(t->o)
    float* aggDo  = aggO   + 24000000L;         // mean(o->d)
    float* aggDdt = aggDo  + 4800000L;          // mean(d->d) + mean(t->d)
    float* c_do   = aggDdt + 4800000L;          // 500000
    float* c_to   = c_do   + 500000L;           // 500000
    float* c_od   = c_to   + 500000L;           // 100000
    float* c_dd   = c_od   + 100000L;           // 100000
    float* c_td   = c_dd   + 100000L;           // 100000

    // zero aggO..c_td (contiguous): 24M + 4.8M + 4.8M + 1.3M floats
    const long nzero = 24000000L + 4800000L + 4800000L + 1300000L;
    k_zero<<<(int)((nzero + 255) / 256), 256, 0, stream>>>(aggO, nzero);

    // encoders
    k_enc<<<(N_ORDER  * H48 + 255) / 256, 256, 0, stream>>>(x_order,  Wo, bo, ho, N_ORDER,  5);
    k_enc<<<(N_DEVICE * H48 + 255) / 256, 256, 0, stream>>>(x_device, Wd, bd, hd, N_DEVICE, 6);
    k_enc<<<(N_TYPE   * H48 + 255) / 256, 256, 0, stream>>>(x_type,   Wt, bt, ht, N_TYPE,   1);

    // per-relation destination counts
    k_count<<<(E_do + 255) / 256, 256, 0, stream>>>(ei_do, E_do, c_do);
    k_count<<<(E_to + 255) / 256, 256, 0, stream>>>(ei_to, E_to, c_to);
    k_count<<<(E_od + 255) / 256, 256, 0, stream>>>(ei_od, E_od, c_od);
    k_count<<<(E_dd + 255) / 256, 256, 0, stream>>>(ei_dd, E_dd, c_dd);
    k_count<<<(E_td + 255) / 256, 256, 0, stream>>>(ei_td, E_td, c_td);
    k_inv<<<(N_ORDER  + 255) / 256, 256, 0, stream>>>(c_do, N_ORDER);
    k_inv<<<(N_ORDER  + 255) / 256, 256, 0, stream>>>(c_to, N_ORDER);
    k_inv<<<(N_DEVICE + 255) / 256, 256, 0, stream>>>(c_od, N_DEVICE);
    k_inv<<<(N_DEVICE + 255) / 256, 256, 0, stream>>>(c_dd, N_DEVICE);
    k_inv<<<(N_DEVICE + 255) / 256, 256, 0, stream>>>(c_td, N_DEVICE);

    // weighted scatters (mean aggregation); relations sharing a concat slot
    // accumulate into the same buffer
    k_scatter<<<(int)(((long)E_do * H48 + 255) / 256), 256, 0, stream>>>(hd, ei_do, c_do, aggO,   E_do);
    k_scatter<<<(int)(((long)E_to * H48 + 255) / 256), 256, 0, stream>>>(ht, ei_to, c_to, aggO,   E_to);
    k_scatter<<<(int)(((long)E_od * H48 + 255) / 256), 256, 0, stream>>>(ho, ei_od, c_od, aggDo,  E_od);
    k_scatter<<<(int)(((long)E_dd * H48 + 255) / 256), 256, 0, stream>>>(hd, ei_dd, c_dd, aggDdt, E_dd);
    k_scatter<<<(int)(((long)E_td * H48 + 255) / 256), 256, 0, stream>>>(ht, ei_td, c_td, aggDdt, E_td);

    // fused update MLP + LN (WMMA f32 16x16x4)
    float* out_o = (float*)d_out;
    float* out_d = out_o + (long)N_ORDER * H48;
    k_update<96><<<(N_ORDER  + 47) / 48, 96, 0, stream>>>(ho, aggO, aggO,   Wuo, buo, g_o, be_o, out_o, N_ORDER);
    k_update<144><<<(N_DEVICE + 47) / 48, 96, 0, stream>>>(hd, aggDo, aggDdt, Wud, bud, g_d, be_d, out_d, N_DEVICE);
}